// Dimer2D_70592082477196
// MI455X (gfx1250) — compile-verified
//
#include <hip/hip_runtime.h>
#include <math.h>

typedef __attribute__((ext_vector_type(2))) float v2f;
typedef __attribute__((ext_vector_type(8))) float v8f;

#define NTHREADS 512
#define NWAVES   (NTHREADS / 32)
#define KSTEPS   100
#define DIMW     8192
#define DIMG     4096

// LDS layout (float offsets)
#define OFF_AS    0        // 2*64*64 = 8192   symmetrized As
#define OFF_V     8192     // 8192             current Lanczos vector
#define OFF_VP    16384    // 8192             previous Lanczos vector
#define OFF_W     24576    // 8192             w = M v
#define OFF_U     32768    // 3*4096 = 12288   GEMM intermediates
#define OFF_RED   45056    // 32               reduction scratch
#define OFF_ALPHA 45088    // 128
#define OFF_BETA  45216    // 128
#define SMEM_FLOATS 45344

// ---------------------------------------------------------------------------
// Wave-cooperative 16x16 tile of a 64x64 GEMM using V_WMMA_F32_16X16X4_F32.
// A row-major (lda), B row-major (ldb). Accumulates into acc.
// ISA layouts (CDNA5 ISA §7.12.2):
//   A 16x4 : lanes 0-15 M=0..15; VGPR0 = K={0|2}, VGPR1 = K={1|3} by lane half
//   B 4x16 : lanes 0-15 N=0..15; VGPR0 = K={0|2}, VGPR1 = K={1|3} by lane half
//   C/D    : VGPR r -> M=r (lanes 0-15) / M=8+r (lanes 16-31), N = lane&15
// ---------------------------------------------------------------------------
__device__ __forceinline__ v8f gemm64_acc(const float* __restrict__ A, int lda,
                                          const float* __restrict__ B, int ldb,
                                          int tm, int tn, int lane, v8f acc) {
  const int half = lane >> 4;
  const int l15  = lane & 15;
  const float* arow = A + (tm + l15) * lda;
  const float* bcol = B + (tn + l15);
#pragma unroll
  for (int k = 0; k < 64; k += 4) {
    const int ka = k + 2 * half;
    v2f a, b;
    a.x = arow[ka];
    a.y = arow[ka + 1];
    b.x = bcol[ka * ldb];
    b.y = bcol[(ka + 1) * ldb];
    // (neg_a, A, neg_b, B, c_mod, C, reuse_a, reuse_b)
    acc = __builtin_amdgcn_wmma_f32_16x16x4_f32(false, a, false, b,
                                                (short)0, acc, false, false);
  }
  return acc;
}

__device__ __forceinline__ void store_tile(float* __restrict__ C, int ldc,
                                           int tm, int tn, int lane, v8f acc) {
  const int half = lane >> 4;
  const int l15  = lane & 15;
#pragma unroll
  for (int r = 0; r < 8; ++r)
    C[(tm + half * 8 + r) * ldc + tn + l15] = acc[r];
}

// Block-wide sum over 512 threads (16 wave32 waves).
__device__ __forceinline__ float block_reduce_sum(float val, float* sRed, int tid) {
  __syncthreads();  // protect sRed reuse from previous call
#pragma unroll
  for (int off = 16; off > 0; off >>= 1)
    val += __shfl_down(val, off, 32);
  const int wid = tid >> 5;
  if ((tid & 31) == 0) sRed[wid] = val;
  __syncthreads();
  if (tid == 0) {
    float s = 0.f;
    for (int i = 0; i < NWAVES; ++i) s += sRed[i];
    sRed[0] = s;
  }
  __syncthreads();
  return sRed[0];
}

// ---------------------------------------------------------------------------
// Structured matvecs.  v layout for Wang: v[l,j,n] -> sv[(l*2+j)*64+n].
// Wang y[k,i,m]: i=0: As0@U(0,1) + As0@U(1,0) + As1@U(0,0); i=1: As0@U(0,0)
// (U(p,j) = V_j @ As[p]); only 3 U combos needed from T's nonzeros.
// ---------------------------------------------------------------------------
__device__ __forceinline__ void matvec_wang(const float* sAs, const float* sv,
                                            float* sU, float* sw, int tid) {
  const int wave = tid >> 5, lane = tid & 31;
  const int tm = (wave >> 2) * 16, tn = (wave & 3) * 16;
  // U slots: 0:(p=0,j=1)  1:(p=1,j=0)  2:(p=0,j=0)
  {
    v8f acc0 = {}, acc1 = {}, acc2 = {};
    acc0 = gemm64_acc(sv + 1 * 64, 128, sAs + 0 * 4096, 64, tm, tn, lane, acc0);
    acc1 = gemm64_acc(sv + 0 * 64, 128, sAs + 1 * 4096, 64, tm, tn, lane, acc1);
    acc2 = gemm64_acc(sv + 0 * 64, 128, sAs + 0 * 4096, 64, tm, tn, lane, acc2);
    store_tile(sU + 0 * 4096, 64, tm, tn, lane, acc0);
    store_tile(sU + 1 * 4096, 64, tm, tn, lane, acc1);
    store_tile(sU + 2 * 4096, 64, tm, tn, lane, acc2);
  }
  __syncthreads();
  {
    v8f acc = {};  // i = 0 plane, accumulate three GEMMs in registers
    acc = gemm64_acc(sAs + 0 * 4096, 64, sU + 0 * 4096, 64, tm, tn, lane, acc);
    acc = gemm64_acc(sAs + 0 * 4096, 64, sU + 1 * 4096, 64, tm, tn, lane, acc);
    acc = gemm64_acc(sAs + 1 * 4096, 64, sU + 2 * 4096, 64, tm, tn, lane, acc);
    store_tile(sw + 0, 128, tm, tn, lane, acc);   // y[k,0,m] at 128*k + m
    v8f acc2 = {};  // i = 1 plane
    acc2 = gemm64_acc(sAs + 0 * 4096, 64, sU + 2 * 4096, 64, tm, tn, lane, acc2);
    store_tile(sw + 64, 128, tm, tn, lane, acc2); // y[k,1,m] at 128*k + 64 + m
  }
  __syncthreads();
}

// Gong y = Sum_o As[o] @ (V @ As[o]);  v is 64x64 row-major.
__device__ __forceinline__ void matvec_gong(const float* sAs, const float* sv,
                                            float* sU, float* sw, int tid) {
  const int wave = tid >> 5, lane = tid & 31;
  const int tm = (wave >> 2) * 16, tn = (wave & 3) * 16;
  {
    v8f x0 = {}, x1 = {};
    x0 = gemm64_acc(sv, 64, sAs + 0 * 4096, 64, tm, tn, lane, x0);
    x1 = gemm64_acc(sv, 64, sAs + 1 * 4096, 64, tm, tn, lane, x1);
    store_tile(sU + 0 * 4096, 64, tm, tn, lane, x0);
    store_tile(sU + 1 * 4096, 64, tm, tn, lane, x1);
  }
  __syncthreads();
  {
    v8f acc = {};
    acc = gemm64_acc(sAs + 0 * 4096, 64, sU + 0 * 4096, 64, tm, tn, lane, acc);
    acc = gemm64_acc(sAs + 1 * 4096, 64, sU + 1 * 4096, 64, tm, tn, lane, acc);
    store_tile(sw, 64, tm, tn, lane, acc);
  }
  __syncthreads();
}

// ---------------------------------------------------------------------------
// k-step Lanczos on the structured operator; returns lambda_max of the
// resulting tridiagonal via Sturm-sequence bisection (== converged Ritz value).
// ---------------------------------------------------------------------------
__device__ float lanczos_lambda(const float* sAs, float* sV, float* sVp, float* sW,
                                float* sU, float* sRed, float* alphas, float* betas,
                                int dim, int which, int tid) {
  // Deterministic pseudo-random start vector.
  for (int i = tid; i < dim; i += NTHREADS) {
    unsigned h = (unsigned)i * 2654435761u + 0x9E3779B9u;
    h ^= h >> 16; h *= 0x85EBCA6Bu; h ^= h >> 13; h *= 0xC2B2AE35u; h ^= h >> 16;
    sV[i]  = (float)h * (1.0f / 2147483648.0f) - 1.0f;
    sVp[i] = 0.0f;
  }
  __syncthreads();
  float p = 0.f;
  for (int i = tid; i < dim; i += NTHREADS) p += sV[i] * sV[i];
  const float n2 = block_reduce_sum(p, sRed, tid);
  const float inv0 = 1.0f / sqrtf(n2);
  for (int i = tid; i < dim; i += NTHREADS) sV[i] *= inv0;
  __syncthreads();

  float beta = 0.0f;
  for (int step = 0; step < KSTEPS; ++step) {
    if (which == 0) matvec_wang(sAs, sV, sU, sW, tid);
    else            matvec_gong(sAs, sV, sU, sW, tid);

    float pa = 0.f;
    for (int i = tid; i < dim; i += NTHREADS) pa += sV[i] * sW[i];
    const float alpha = block_reduce_sum(pa, sRed, tid);

    float sq = 0.f;
    for (int i = tid; i < dim; i += NTHREADS) {
      const float w = sW[i] - alpha * sV[i] - beta * sVp[i];
      sW[i] = w;
      sq += w * w;
    }
    const float beta_new = sqrtf(block_reduce_sum(sq, sRed, tid));
    const float invb = 1.0f / fmaxf(beta_new, 1e-30f);
    for (int i = tid; i < dim; i += NTHREADS) {
      const float w = sW[i];
      sVp[i] = sV[i];
      sV[i]  = w * invb;
    }
    __syncthreads();
    if (tid == 0) { alphas[step] = alpha; betas[step] = beta_new; }
    beta = beta_new;
    __syncthreads();
  }

  // lambda_max of tridiag(alphas; betas[0..K-2]) via Sturm bisection (thread 0).
  if (tid == 0) {
    float lo = 1e30f, hi = -1e30f;
    for (int i = 0; i < KSTEPS; ++i) {
      const float bl = (i > 0)          ? fabsf(betas[i - 1]) : 0.f;
      const float br = (i < KSTEPS - 1) ? fabsf(betas[i])     : 0.f;
      lo = fminf(lo, alphas[i] - bl - br);
      hi = fmaxf(hi, alphas[i] + bl + br);
    }
    for (int it = 0; it < 64; ++it) {
      const float x = 0.5f * (lo + hi);
      int cnt = 0;
      float d = alphas[0] - x;
      if (d < 0.f) cnt++;
      for (int i = 1; i < KSTEPS; ++i) {
        float dd = d;
        if (fabsf(dd) < 1e-25f) dd = (dd < 0.f) ? -1e-25f : 1e-25f;
        const float bi = betas[i - 1];
        d = (alphas[i] - x) - bi * bi / dd;
        if (d < 0.f) cnt++;
      }
      if (cnt >= KSTEPS) hi = x; else lo = x;
    }
    sRed[0] = 0.5f * (lo + hi);
  }
  __syncthreads();
  const float lam = sRed[0];
  __syncthreads();
  return lam;
}

__global__ __launch_bounds__(NTHREADS)
void dimer2d_kernel(const float* __restrict__ A, float* __restrict__ out) {
  extern __shared__ float smem[];
  float* sAs    = smem + OFF_AS;
  float* sV     = smem + OFF_V;
  float* sVp    = smem + OFF_VP;
  float* sW     = smem + OFF_W;
  float* sU     = smem + OFF_U;
  float* sRed   = smem + OFF_RED;
  float* alphas = smem + OFF_ALPHA;
  float* betas  = smem + OFF_BETA;
  const int tid = threadIdx.x;

  // Symmetrize A into LDS: As[o] = 0.5*(A[o] + A[o]^T)
  for (int idx = tid; idx < 2 * 64 * 64; idx += NTHREADS) {
    const int o = idx >> 12, r = idx & 4095;
    const int k = r >> 6, l = r & 63;
    sAs[idx] = 0.5f * (A[o * 4096 + k * 64 + l] + A[o * 4096 + l * 64 + k]);
  }
  __syncthreads();

  const float lamW = lanczos_lambda(sAs, sV, sVp, sW, sU, sRed, alphas, betas,
                                    DIMW, /*which=*/0, tid);
  const float lamG = lanczos_lambda(sAs, sV, sVp, sW, sU, sRed, alphas, betas,
                                    DIMG, /*which=*/1, tid);
  if (tid == 0) out[0] = logf(lamW / lamG);
}

extern "C" void kernel_launch(void* const* d_in, const int* in_sizes, int n_in,
                              void* d_out, int out_size, void* d_ws, size_t ws_size,
                              hipStream_t stream) {
  (void)in_sizes; (void)n_in; (void)out_size; (void)d_ws; (void)ws_size;
  const float* A = (const float*)d_in[0];
  float* out = (float*)d_out;
  const size_t shmem = (size_t)SMEM_FLOATS * sizeof(float);  // ~181 KB (<320 KB WGP LDS)
  (void)hipFuncSetAttribute(reinterpret_cast<const void*>(&dimer2d_kernel),
                            hipFuncAttributeMaxDynamicSharedMemorySize, (int)shmem);
  dimer2d_kernel<<<dim3(1), dim3(NTHREADS), shmem, stream>>>(A, out);
}